// Blurring_Model_73701638799598
// MI455X (gfx1250) — compile-verified
//
#include <hip/hip_runtime.h>
#include <math.h>
#include <stdint.h>

typedef __attribute__((ext_vector_type(2))) float v2f;
typedef __attribute__((ext_vector_type(8))) float v8f;

// Tile geometry: each workgroup produces a 16(x) x 16(y) x 8(z) output brick.
#define TX 16
#define TY 16
#define TZ 8
#define PADR 4
#define INX 24            // TX + 2*PADR
#define INY 24            // TY + 2*PADR
#define INZ 16            // TZ + 2*PADR
// LDS float offsets
#define OFF_IN   0        // sIn  [16][24][24] = 9216 floats
#define OFF_T1   9216     // sT1  [ 8][24][24] = 4608 floats
#define OFF_WX   13824    // sWx  [9]
#define SMEM_FLOATS 13840 // padded
// sT2 [8][16][24] = 3072 floats aliases sIn (raw input dead after z-pass).

__global__ __launch_bounds__(256)
void gauss3d_fused_kernel(const float* __restrict__ xin,
                          const float* __restrict__ psx,
                          const float* __restrict__ psy,
                          const float* __restrict__ psz,
                          float* __restrict__ out)
{
    extern __shared__ float smem[];
    float* sIn = smem + OFF_IN;
    float* sT1 = smem + OFF_T1;
    float* sWx = smem + OFF_WX;
    float* sT2 = smem + OFF_IN;   // alias (safe: barrier-separated phases)

    const int tid = threadIdx.x;
    const int x0 = blockIdx.x * TX;
    const int y0 = blockIdx.y * TY;
    const int z0 = (blockIdx.z & 15) * TZ;
    const int b  = blockIdx.z >> 4;
    const int inBase = b * 2097152; // 128^3

    // ---- Phase 1: stage the 24x24x16 halo tile into LDS ----
    // Interior tiles (no zero padding needed): async global->LDS B128 DMA,
    // 16B-aligned rows, no VGPR staging; overlaps with weight computation.
    const bool interior = (x0 >= 16 && x0 <= 96) &&
                          (y0 >= 16 && y0 <= 96) &&
                          (z0 >=  8 && z0 <= 112);
    if (interior) {
        const float* gbase = xin + (inBase + (z0 - 4) * 16384
                                           + (y0 - 4) * 128 + (x0 - 4));
        const unsigned ldsBase = (unsigned)(uintptr_t)sIn;
        for (int i = tid; i < INZ * INY * (INX / 4); i += 256) {  // 2304 vec4s
            int tx4 = i % 6;
            int ty  = (i / 6) % INY;
            int tz  = i / (6 * INY);
            unsigned long long ga =
                (unsigned long long)(uintptr_t)(gbase + tz * 16384 + ty * 128 + tx4 * 4);
            unsigned lo = ldsBase + (unsigned)(i * 16);
            asm volatile("global_load_async_to_lds_b128 %0, %1, off"
                         :: "v"(lo), "v"(ga) : "memory");
        }
    } else {
        // Edge tiles: element-wise with zero padding outside the volume.
        for (int i = tid; i < INX * INY * INZ; i += 256) {
            int tx  = i % INX;
            int rem = i / INX;
            int ty  = rem % INY;
            int tz  = rem / INY;
            int gx = x0 - PADR + tx;
            int gy = y0 - PADR + ty;
            int gz = z0 - PADR + tz;
            float v = 0.f;
            if ((unsigned)gx < 128u && (unsigned)gy < 128u && (unsigned)gz < 128u)
                v = xin[inBase + gz * 16384 + gy * 128 + gx];
            sIn[i] = v;
        }
    }

    // ---- per-axis normalized 1D Gaussian weights (spacing = 4.0) ----
    // (computed while the async DMA is in flight)
    const float ssx = psx[0] * 0.25f;
    const float ssy = psy[0] * 0.25f;
    const float ssz = psz[0] * 0.25f;
    float wy9[9], wz9[9];
    {
        float ey[9], ez[9];
        float sy = 0.f, sz = 0.f;
        const float iy = -0.5f / (ssy * ssy);
        const float iz = -0.5f / (ssz * ssz);
        #pragma unroll
        for (int t = 0; t < 9; ++t) {
            float d = (float)t - 4.0f;
            ey[t] = expf(d * d * iy); sy += ey[t];
            ez[t] = expf(d * d * iz); sz += ez[t];
        }
        float ry = 1.f / sy, rz = 1.f / sz;
        #pragma unroll
        for (int t = 0; t < 9; ++t) { wy9[t] = ey[t] * ry; wz9[t] = ez[t] * rz; }
    }
    if (tid == 0) {
        float ex[9], s = 0.f;
        const float ix = -0.5f / (ssx * ssx);
        #pragma unroll
        for (int t = 0; t < 9; ++t) {
            float d = (float)t - 4.0f;
            ex[t] = expf(d * d * ix); s += ex[t];
        }
        float r = 1.f / s;
        #pragma unroll
        for (int t = 0; t < 9; ++t) sWx[t] = ex[t] * r;
    }

    if (interior) {
        asm volatile("s_wait_asynccnt 0x0" ::: "memory");
    }
    __syncthreads();

    // ---- Phase 2: z-convolution (VALU, float4 LDS traffic) ----
    // sT1[zo][y][x] = sum_t wz[t] * sIn[zo+t][y][x] ; 8*24*6 = 1152 float4s
    {
        const float4* sIn4 = (const float4*)sIn;
        float4* sT14 = (float4*)sT1;
        for (int i = tid; i < TZ * INY * (INX / 4); i += 256) {
            int xv = i % 6;
            int yy = (i / 6) % INY;
            int zo = i / (6 * INY);
            float4 acc = make_float4(0.f, 0.f, 0.f, 0.f);
            #pragma unroll
            for (int t = 0; t < 9; ++t) {
                float4 s = sIn4[(zo + t) * 144 + yy * 6 + xv];
                float w = wz9[t];
                acc.x += w * s.x; acc.y += w * s.y;
                acc.z += w * s.z; acc.w += w * s.w;
            }
            sT14[zo * 144 + yy * 6 + xv] = acc;
        }
    }
    __syncthreads();

    // ---- Phase 3: y-convolution (VALU) ----
    // sT2[z][yo][x] = sum_t wy[t] * sT1[z][yo+t][x] ; 8*16*6 = 768 float4s
    {
        const float4* sT14 = (const float4*)sT1;
        float4* sT24 = (float4*)sT2;
        for (int i = tid; i < TZ * TY * (INX / 4); i += 256) {
            int xv = i % 6;
            int yo = (i / 6) % TY;
            int z  = i / (6 * TY);
            float4 acc = make_float4(0.f, 0.f, 0.f, 0.f);
            #pragma unroll
            for (int t = 0; t < 9; ++t) {
                float4 s = sT14[z * 144 + (yo + t) * 6 + xv];
                float w = wy9[t];
                acc.x += w * s.x; acc.y += w * s.y;
                acc.z += w * s.z; acc.w += w * s.w;
            }
            sT24[z * 96 + yo * 6 + xv] = acc;
        }
    }
    __syncthreads();

    // ---- Phase 4: x-convolution as banded matmul on the matrix pipe ----
    // out(16y x 16x) = InWindow(16 x 24) * W(24 x 16), W[j][n] = wx[j-n] on band.
    // Six chained V_WMMA_F32_16X16X4_F32, K split 24 = 6*4. One z-slice per wave.
    {
        const int lane = tid & 31;
        const int wave = tid >> 5;
        const int r  = lane & 15;   // A-matrix row (M) / D-matrix column (N)
        const int kh = lane >> 4;   // K-half select per ISA 16x4 fp32 layout

        // Build B operands once: B[kk][n] with kk = 2*kh, 2*kh+1 per lane.
        v2f bw[6];
        #pragma unroll
        for (int c = 0; c < 6; ++c) {
            int j0 = 4 * c + 2 * kh;
            int u0 = j0 - r, u1 = j0 + 1 - r;
            float b0 = (u0 >= 0 && u0 <= 8) ? sWx[u0] : 0.f;
            float b1 = (u1 >= 0 && u1 <= 8) ? sWx[u1] : 0.f;
            bw[c].x = b0; bw[c].y = b1;
        }

        const int z = wave;                       // 8 waves -> 8 z-slices
        const float* row = &sT2[z * 384 + r * 24];
        v8f acc = {0.f, 0.f, 0.f, 0.f, 0.f, 0.f, 0.f, 0.f};
        #pragma unroll
        for (int c = 0; c < 6; ++c) {
            int j0 = 4 * c + 2 * kh;
            v2f a = *(const v2f*)(row + j0);      // 8B-aligned LDS load
            acc = __builtin_amdgcn_wmma_f32_16x16x4_f32(
                      false, a, false, bw[c], (short)0, acc, false, false);
        }

        // D layout: VGPR v -> M = v + 8*kh (lanes>=16), N = lane&15.
        const int outXY = inBase + (z0 + z) * 16384 + x0 + r;
        const int mbase = y0 + kh * 8;
        #pragma unroll
        for (int v = 0; v < 8; ++v)
            out[outXY + (mbase + v) * 128] = acc[v];
    }
}

extern "C" void kernel_launch(void* const* d_in, const int* in_sizes, int n_in,
                              void* d_out, int out_size, void* d_ws, size_t ws_size,
                              hipStream_t stream) {
    const float* x   = (const float*)d_in[0];
    const float* ssx = (const float*)d_in[1];
    const float* ssy = (const float*)d_in[2];
    const float* ssz = (const float*)d_in[3];
    float* out = (float*)d_out;

    dim3 grid(128 / TX, 128 / TY, (128 / TZ) * 8);   // x-tiles, y-tiles, z-tiles*batch
    dim3 block(256);
    size_t smem = SMEM_FLOATS * sizeof(float);        // 55,360 B < 64 KB
    hipLaunchKernelGGL(gauss3d_fused_kernel, grid, block, smem, stream,
                       x, ssx, ssy, ssz, out);
}